// ResidualNetwork_627065225937
// MI455X (gfx1250) — compile-verified
//
#include <hip/hip_runtime.h>
#include <hip/hip_bf16.h>

// ---------------------------------------------------------------------------
// ResidualNetwork on gfx1250 (MI455X): LN -> exchange(w1) -> exchange(w2) -> residual
// GEMMs run on V_WMMA_F32_16X16X4_F32 (fp32 matrix pipe, exact precision).
// Segment means are projected per-segment (50k x 64 @ 64x64) BEFORE the gather,
// collapsing the reference's K=256 GEMM to K=64 + two cached gathers + a const.
// ---------------------------------------------------------------------------

typedef __attribute__((ext_vector_type(2))) float v2f;
typedef __attribute__((ext_vector_type(8))) float v8f;

#define NROWS 500000
#define NF 64
#define NSEG 50000
#define EPSV 1e-5f
#define SLOPEV 0.01f

__device__ __forceinline__ float leaky_f(float x) { return x >= 0.f ? x : SLOPEV * x; }

__device__ __forceinline__ void atomAddF(float* p, float v) {
    __hip_atomic_fetch_add(p, v, __ATOMIC_RELAXED, __HIP_MEMORY_SCOPE_AGENT);
}

// 16x64 (A, fragment layout) @ 64x64 row-major (w) accumulated into 4 N-tiles.
// A fragment (32-bit A-matrix 16x4): lanes 0-15 hold M=0..15 with K={0,1} of each
// 4-wide K-step; lanes 16-31 hold K={2,3}. B fragment mirrors it: per lane,
// b = { w[k0][col], w[k0+1][col] } with k0 = 4*ks + 2*half, col = nt*16 + (lane&15).
__device__ __forceinline__ void gemm_tile_16x64(const v2f a[16],
                                                const float* __restrict__ w,
                                                v8f acc[4], int n, int half) {
#pragma unroll
    for (int nt = 0; nt < 4; ++nt) {
        v8f c = acc[nt];
#pragma unroll
        for (int ks = 0; ks < 16; ++ks) {
            const int k0 = 4 * ks + 2 * half;
            v2f b;
            b.x = w[(k0    ) * NF + nt * 16 + n];
            b.y = w[(k0 + 1) * NF + nt * 16 + n];
            c = __builtin_amdgcn_wmma_f32_16x16x4_f32(false, a[ks], false, b,
                                                      (short)0, c, false, false);
        }
        acc[nt] = c;
    }
}

// ---- Kernel 1: LayerNorm + scatter normed rows into segment sums (1 wave/row) ----
__global__ __launch_bounds__(256) void k_ln_scatter(
    const float* __restrict__ x, const int* __restrict__ idx,
    const float* __restrict__ gamma, const float* __restrict__ beta,
    float* rowsum, float* colsum, float* cntr, float* cntc) {
    const int wave = (blockIdx.x * blockDim.x + threadIdx.x) >> 5;
    const int lane = threadIdx.x & 31;
    if (wave >= NROWS) return;
    const int r = wave;
    v2f xv = *(const v2f*)(x + (size_t)r * NF + 2 * lane);
    float s = xv.x + xv.y;
    float q = xv.x * xv.x + xv.y * xv.y;
#pragma unroll
    for (int m = 16; m >= 1; m >>= 1) {
        s += __shfl_xor(s, m, 32);
        q += __shfl_xor(q, m, 32);
    }
    const float mean = s * (1.f / NF);
    const float rstd = rsqrtf(q * (1.f / NF) - mean * mean + EPSV);
    v2f g = *(const v2f*)(gamma + 2 * lane);
    v2f b = *(const v2f*)(beta + 2 * lane);
    const float n0 = (xv.x - mean) * rstd * g.x + b.x;
    const float n1 = (xv.y - mean) * rstd * g.y + b.y;
    const int rr = idx[2 * r], cc = idx[2 * r + 1];
    float* pr = rowsum + (size_t)rr * NF + 2 * lane;
    float* pc = colsum + (size_t)cc * NF + 2 * lane;
    atomAddF(pr, n0);     atomAddF(pr + 1, n1);
    atomAddF(pc, n0);     atomAddF(pc + 1, n1);
    if (lane == 0) { atomAddF(cntr + rr, 1.f); atomAddF(cntc + cc, 1.f); }
}

// ---- Column-sum of a NSEGxNF buffer -> glob[64] (for the global-mean term) ----
__global__ __launch_bounds__(256) void k_colsum_reduce(const float* __restrict__ sums,
                                                       float* __restrict__ glob) {
    const int c = threadIdx.x & 63;
    const int grp = threadIdx.x >> 6;  // 0..3
    const int SPB = (NSEG + gridDim.x - 1) / gridDim.x;
    const int s0 = blockIdx.x * SPB;
    const int s1 = min(s0 + SPB, NSEG);
    float acc = 0.f;
    for (int s = s0 + grp; s < s1; s += 4) acc += sums[(size_t)s * NF + c];
    __shared__ float lds[256];
    lds[threadIdx.x] = acc;
    __syncthreads();
    if (grp == 0) {
        float t = lds[c] + lds[64 + c] + lds[128 + c] + lds[192 + c];
        atomAddF(glob + c, t);
    }
}

// ---- Kernel 2: in-place segment mean -> projected segment vector (WMMA) ----
__global__ __launch_bounds__(256) void k_seg_project(float* __restrict__ sums,
                                                     const float* __restrict__ cnt,
                                                     const float* __restrict__ w) {
    const int wave = (blockIdx.x * blockDim.x + threadIdx.x) >> 5;
    const int lane = threadIdx.x & 31;
    if (wave >= NSEG / 16) return;
    const int segbase = wave * 16;
    const int n = lane & 15, half = lane >> 4;
    const int seg = segbase + n;
    const float inv = 1.f / fmaxf(cnt[seg], 1.f);
    v2f a[16];
#pragma unroll
    for (int ks = 0; ks < 16; ++ks) {
        v2f t = *(const v2f*)(sums + (size_t)seg * NF + 4 * ks + 2 * half);
        t.x *= inv; t.y *= inv;
        a[ks] = t;
    }
    v8f acc[4] = {};
    gemm_tile_16x64(a, w, acc, n, half);
#pragma unroll
    for (int v = 0; v < 8; ++v) {
        const int m = v + 8 * half;
#pragma unroll
        for (int nt = 0; nt < 4; ++nt)
            sums[(size_t)(segbase + m) * NF + nt * 16 + n] = acc[nt][v];
    }
}

// ---- cvec[n] = (glob/N) @ w_glob + bias  (single 64-thread block) ----
__global__ void k_cvec(const float* __restrict__ glob, const float* __restrict__ w,
                       const float* __restrict__ bias, float* __restrict__ cvec) {
    const int nn = threadIdx.x;
    float acc = bias[nn];
    const float invN = 1.f / (float)NROWS;
#pragma unroll 8
    for (int k = 0; k < NF; ++k) acc += glob[k] * invN * w[k * NF + nn];
    cvec[nn] = acc;
}

// ---- Kernel 3: LN(x) @ w1_x + gathers + leaky -> step1, fused scatter for exch2 ----
__global__ __launch_bounds__(256) void k_gemm1(
    const float* __restrict__ x, const int* __restrict__ idx,
    const float* __restrict__ gamma, const float* __restrict__ beta,
    const float* __restrict__ w1x, const float* __restrict__ rowproj,
    const float* __restrict__ colproj, const float* __restrict__ cvec,
    float* __restrict__ step1, float* rowsum2, float* colsum2) {
    const int wave = (blockIdx.x * blockDim.x + threadIdx.x) >> 5;
    const int lane = threadIdx.x & 31;
    if (wave >= NROWS / 16) return;
    const int R = wave * 16;
    const int n = lane & 15, half = lane >> 4;
    const int arow = R + n;
    v2f a[16];
    float s = 0.f, q = 0.f;
#pragma unroll
    for (int ks = 0; ks < 16; ++ks) {
        v2f t = *(const v2f*)(x + (size_t)arow * NF + 4 * ks + 2 * half);
        a[ks] = t;
        s += t.x + t.y;
        q += t.x * t.x + t.y * t.y;
    }
    s += __shfl_xor(s, 16, 32);   // row 'arow' lives in lanes (n) and (n+16)
    q += __shfl_xor(q, 16, 32);
    const float mean = s * (1.f / NF);
    const float rstd = rsqrtf(q * (1.f / NF) - mean * mean + EPSV);
#pragma unroll
    for (int ks = 0; ks < 16; ++ks) {
        v2f g = *(const v2f*)(gamma + 4 * ks + 2 * half);
        v2f b = *(const v2f*)(beta + 4 * ks + 2 * half);
        a[ks].x = (a[ks].x - mean) * rstd * g.x + b.x;
        a[ks].y = (a[ks].y - mean) * rstd * g.y + b.y;
    }
    v8f acc[4] = {};
    gemm_tile_16x64(a, w1x, acc, n, half);
#pragma unroll
    for (int v = 0; v < 8; ++v) {
        const int m = v + 8 * half;
        const int row = R + m;
        const int rid = idx[2 * row], cid = idx[2 * row + 1];
#pragma unroll
        for (int nt = 0; nt < 4; ++nt) {
            const int col = nt * 16 + n;
            float val = acc[nt][v]
                      + rowproj[(size_t)rid * NF + col]
                      + colproj[(size_t)cid * NF + col]
                      + cvec[col];
            val = leaky_f(val);
            step1[(size_t)row * NF + col] = val;
            atomAddF(rowsum2 + (size_t)rid * NF + col, val);
            atomAddF(colsum2 + (size_t)cid * NF + col, val);
        }
    }
}

// ---- Kernel 5: step1 @ w2_x + gathers + leaky, residual + final leaky -> out ----
__global__ __launch_bounds__(256) void k_gemm2(
    const float* __restrict__ step1, const float* __restrict__ x0,
    const int* __restrict__ idx, const float* __restrict__ w2x,
    const float* __restrict__ rowproj, const float* __restrict__ colproj,
    const float* __restrict__ cvec, float* __restrict__ out) {
    const int wave = (blockIdx.x * blockDim.x + threadIdx.x) >> 5;
    const int lane = threadIdx.x & 31;
    if (wave >= NROWS / 16) return;
    const int R = wave * 16;
    const int n = lane & 15, half = lane >> 4;
    const int arow = R + n;
    v2f a[16];
#pragma unroll
    for (int ks = 0; ks < 16; ++ks)
        a[ks] = *(const v2f*)(step1 + (size_t)arow * NF + 4 * ks + 2 * half);
    v8f acc[4] = {};
    gemm_tile_16x64(a, w2x, acc, n, half);
#pragma unroll
    for (int v = 0; v < 8; ++v) {
        const int m = v + 8 * half;
        const int row = R + m;
        const int rid = idx[2 * row], cid = idx[2 * row + 1];
#pragma unroll
        for (int nt = 0; nt < 4; ++nt) {
            const int col = nt * 16 + n;
            float val = acc[nt][v]
                      + rowproj[(size_t)rid * NF + col]
                      + colproj[(size_t)cid * NF + col]
                      + cvec[col];
            val = leaky_f(val);                          // step2
            float o = x0[(size_t)row * NF + col] + val;  // residual
            out[(size_t)row * NF + col] = leaky_f(o);
        }
    }
}

extern "C" void kernel_launch(void* const* d_in, const int* in_sizes, int n_in,
                              void* d_out, int out_size, void* d_ws, size_t ws_size,
                              hipStream_t stream) {
    const float* x     = (const float*)d_in[0];
    const int*   idx   = (const int*)d_in[1];   // (N,2) interleaved row,col
    const float* gamma = (const float*)d_in[2];
    const float* beta  = (const float*)d_in[3];
    const float* w1    = (const float*)d_in[4]; // (256,64) row-major
    const float* b1    = (const float*)d_in[5];
    const float* w2    = (const float*)d_in[6];
    const float* b2    = (const float*)d_in[7];
    float* out = (float*)d_out;

    // Workspace layout (floats). Total ~180 MB.
    float* ws      = (float*)d_ws;
    float* step1   = ws;                                 // N*64
    float* rowsum1 = step1 + (size_t)NROWS * NF;         // NSEG*64
    float* colsum1 = rowsum1 + (size_t)NSEG * NF;
    float* rowsum2 = colsum1 + (size_t)NSEG * NF;
    float* colsum2 = rowsum2 + (size_t)NSEG * NF;
    float* cntr    = colsum2 + (size_t)NSEG * NF;        // NSEG
    float* cntc    = cntr + NSEG;                        // NSEG
    float* glob1   = cntc + NSEG;                        // 64
    float* glob2   = glob1 + NF;                         // 64
    float* cvec1   = glob2 + NF;                         // 64
    float* cvec2   = cvec1 + NF;                         // 64
    (void)ws_size; (void)in_sizes; (void)n_in; (void)out_size;

    // Zero the accumulation scratch every call (harness poisons ws once).
    const size_t zero_bytes = (4 * (size_t)NSEG * NF + 2 * NSEG + 2 * NF) * sizeof(float);
    hipMemsetAsync(rowsum1, 0, zero_bytes, stream);

    // 1) LN + scatter into exchange-1 segment sums (+counts). 1 wave per row.
    k_ln_scatter<<<(NROWS * 32 + 255) / 256, 256, 0, stream>>>(
        x, idx, gamma, beta, rowsum1, colsum1, cntr, cntc);

    // 2) glob1 = column sums of rowsum1 (total sum of normed rows).
    k_colsum_reduce<<<64, 256, 0, stream>>>(rowsum1, glob1);

    // 3) Project segment means in place (WMMA), w1 blocks 64:128 / 128:192.
    const int projBlocks = ((NSEG / 16) * 32 + 255) / 256;
    k_seg_project<<<projBlocks, 256, 0, stream>>>(rowsum1, cntr, w1 + 64 * NF);
    k_seg_project<<<projBlocks, 256, 0, stream>>>(colsum1, cntc, w1 + 128 * NF);

    // 4) Constant vector: (glob/N) @ w1[192:256] + b1.
    k_cvec<<<1, 64, 0, stream>>>(glob1, w1 + 192 * NF, b1, cvec1);

    // 5) Main GEMM 1 (fused LN recompute, gathers, leaky, exchange-2 scatter).
    const int gemmBlocks = ((NROWS / 16) * 32 + 255) / 256;
    k_gemm1<<<gemmBlocks, 256, 0, stream>>>(
        x, idx, gamma, beta, w1, rowsum1, colsum1, cvec1, step1, rowsum2, colsum2);

    // 6) Exchange 2: glob2, projections, cvec2.
    k_colsum_reduce<<<64, 256, 0, stream>>>(rowsum2, glob2);
    k_seg_project<<<projBlocks, 256, 0, stream>>>(rowsum2, cntr, w2 + 64 * NF);
    k_seg_project<<<projBlocks, 256, 0, stream>>>(colsum2, cntc, w2 + 128 * NF);
    k_cvec<<<1, 64, 0, stream>>>(glob2, w2 + 192 * NF, b2, cvec2);

    // 7) Main GEMM 2 + residual + final leaky -> out.
    k_gemm2<<<gemmBlocks, 256, 0, stream>>>(
        step1, x, idx, w2, rowsum2, colsum2, cvec2, out);
}